// Euclidean_11029476016273
// MI455X (gfx1250) — compile-verified
//
#include <hip/hip_runtime.h>

// Squared Euclidean distance: out[r,u] = |x_r|^2 + |w_u|^2 - 2 x_r . w_u
// fp32 end to end. Output (134 MB) dominates traffic -> memory bound; keep
// waves free for WMMA+store by staging GEMM tiles with the Tensor Data Mover.

typedef __attribute__((ext_vector_type(2))) float v2f;
typedef __attribute__((ext_vector_type(8))) float v8f;
typedef __attribute__((ext_vector_type(4))) unsigned int v4u;
typedef __attribute__((ext_vector_type(4))) int v4i;
typedef __attribute__((ext_vector_type(8))) int v8i;

#define BN_ROWS 32768   // B*N = 8*4096
#define DIMS    128     // D (= K of the GEMM)
#define UNITS   1024    // U

#define MB 64           // output rows per workgroup
#define NB 64           // output cols per workgroup
#define KC 32           // K chunk staged in LDS (4 chunks, double buffered)
#define LDX (KC + 4)    // X tile LDS row stride (36 floats): TDM pad inserts
                        // 4 DWORDs per 32-DWORD row -> conflict-free A reads
#define NCHUNK (DIMS / KC)

// X tile descriptor pad bits: pad_enable(20), pad_interval code 4 = 32 DWORDs
// (22:24), pad_amount code 3 = 4 DWORDs (25:31)
#define XPAD ((1u << 20) | (4u << 22) | (3u << 25))

// ---------------------------------------------------------------------------
// Row norms |x_r|^2
// ---------------------------------------------------------------------------
__global__ __launch_bounds__(256) void row_norms_kernel(
    const float* __restrict__ X, float* __restrict__ x2g) {
  int row = blockIdx.x * 256 + threadIdx.x;
  if (row >= BN_ROWS) return;
  const float4* p = reinterpret_cast<const float4*>(X + (size_t)row * DIMS);
  float s = 0.0f;
#pragma unroll
  for (int i = 0; i < DIMS / 4; ++i) {
    float4 v = p[i];
    s += v.x * v.x + v.y * v.y + v.z * v.z + v.w * v.w;
  }
  x2g[row] = s;
}

// ---------------------------------------------------------------------------
// Column norms |w_u|^2  (w is [D, U] row-major)
// ---------------------------------------------------------------------------
__global__ __launch_bounds__(256) void col_norms_kernel(
    const float* __restrict__ Wm, float* __restrict__ w2g) {
  int u = blockIdx.x * 256 + threadIdx.x;
  if (u >= UNITS) return;
  float s = 0.0f;
#pragma unroll 8
  for (int d = 0; d < DIMS; ++d) {
    float v = Wm[(size_t)d * UNITS + u];
    s += v * v;
  }
  w2g[u] = s;
}

// ---------------------------------------------------------------------------
// TDM: DMA a 2D tile (tile_d1 rows x tile_d0 elements, 4-byte data) from a
// row-major [tensor_d1 x tensor_d0] fp32 tensor into LDS. D# layout per
// CDNA5 ISA ch.8 (group0: flags/lds/global addr; group1: dims/strides/pad).
// ---------------------------------------------------------------------------
__device__ __forceinline__ void tdm_load_2d(
    unsigned lds_addr, unsigned long long gaddr,
    unsigned tensor_d0, unsigned tensor_d1,
    unsigned tile_d0, unsigned tile_d1,
    unsigned d0_stride, unsigned pad_bits) {
  v4u g0;
  g0.x = 1u;                                   // count=1, user descriptor
  g0.y = lds_addr;                             // lds_addr[31:0] (bytes)
  g0.z = (unsigned)gaddr;                      // global_addr[31:0]
  g0.w = (unsigned)(gaddr >> 32) | (2u << 30); // global_addr[56:32] | type=2

  v8i g1;
  g1[0] = (int)((2u << 16) | pad_bits);        // data_size=2 (4B) + pad cfg
  g1[1] = (int)(tensor_d0 << 16);              // tensor_dim0[15:0] @ bits 63:48
  g1[2] = (int)((tensor_d0 >> 16) | (tensor_d1 << 16)); // d0 hi | d1 lo
  g1[3] = (int)((tensor_d1 >> 16) | (tile_d0 << 16));   // d1 hi | tile_dim0
  g1[4] = (int)(tile_d1 & 0xFFFFu);            // tile_dim1 (tile_dim2 = 0)
  g1[5] = (int)d0_stride;                      // tensor_dim0_stride[31:0]
  g1[6] = 0;                                   // stride hi / dim1_stride lo
  g1[7] = 0;                                   // dim1_stride hi (2D: unused)

  v4i gz = {0, 0, 0, 0};                       // groups 2/3: 2-D tensor
#if defined(__clang_major__) && (__clang_major__ >= 23)
  v8i gz8 = {0, 0, 0, 0, 0, 0, 0, 0};
  __builtin_amdgcn_tensor_load_to_lds(g0, g1, gz, gz, gz8, 0);
#else
  __builtin_amdgcn_tensor_load_to_lds(g0, g1, gz, gz, 0);
#endif
}

// ---------------------------------------------------------------------------
// WMMA GEMM + distance epilogue. 128 threads = 4 waves (2x2 over M,N); each
// wave owns a 32x32 tile = 2x2 V_WMMA_F32_16X16X4_F32 accumulators. Tiles are
// staged by the TDM, double buffered, synced with s_wait_tensorcnt + barrier.
// ---------------------------------------------------------------------------
__global__ __launch_bounds__(128) void dist_gemm_kernel(
    const float* __restrict__ X, const float* __restrict__ Wm,
    const float* __restrict__ x2g, const float* __restrict__ w2g,
    float* __restrict__ out) {
  __shared__ float lds_x[2][MB][LDX];  // 2 x 64 x 36 fp32 (TDM-padded rows)
  __shared__ float lds_w[2][KC][NB];   // 2 x 32 x 64 fp32

  const int tid  = threadIdx.x;
  const int lane = tid & 31;
  const int wv   = tid >> 5;   // wave 0..3
  const int wm   = wv >> 1;    // wave M index 0..1
  const int wn   = wv & 1;     // wave N index 0..1
  const int hi   = lane >> 4;  // half-wave select (K+2 / M+8)
  const int ln   = lane & 15;

  const int blockM = blockIdx.y * MB;
  const int blockN = blockIdx.x * NB;

  v8f acc[2][2] = {};

  // Issue chunk 0 DMA (wave 0 only; TDM ignores EXEC, tracked by TENSORcnt).
  if (wv == 0) {
    tdm_load_2d((unsigned)(unsigned long long)&lds_x[0][0][0],
                (unsigned long long)X + ((unsigned long long)blockM * DIMS) * 4ull,
                DIMS, BN_ROWS, KC, MB, DIMS, XPAD);
    tdm_load_2d((unsigned)(unsigned long long)&lds_w[0][0][0],
                (unsigned long long)Wm + ((unsigned long long)blockN) * 4ull,
                UNITS, DIMS, NB, KC, UNITS, 0u);
  }

#pragma unroll
  for (int c = 0; c < NCHUNK; ++c) {
    const int buf = c & 1;
    if (wv == 0) {
      if (c + 1 < NCHUNK) {
        const int k1 = (c + 1) * KC;
        const int nbuf = (c + 1) & 1;
        tdm_load_2d((unsigned)(unsigned long long)&lds_x[nbuf][0][0],
                    (unsigned long long)X +
                        ((unsigned long long)blockM * DIMS + k1) * 4ull,
                    DIMS, BN_ROWS, KC, MB, DIMS, XPAD);
        tdm_load_2d((unsigned)(unsigned long long)&lds_w[nbuf][0][0],
                    (unsigned long long)Wm +
                        ((unsigned long long)k1 * UNITS + blockN) * 4ull,
                    UNITS, DIMS, NB, KC, UNITS, 0u);
        __builtin_amdgcn_s_wait_tensorcnt(2);  // current chunk complete
      } else {
        __builtin_amdgcn_s_wait_tensorcnt(0);  // last chunk complete
      }
    }
    __syncthreads();  // release all waves; LDS tile valid

    // ---- 8 K-steps of 4; each step issues 4 v_wmma_f32_16x16x4_f32 ----
#pragma unroll
    for (int kk = 0; kk < KC; kk += 4) {
      v2f a[2];
#pragma unroll
      for (int mi = 0; mi < 2; ++mi) {
        const float* p = &lds_x[buf][wm * 32 + mi * 16 + ln][kk + 2 * hi];
        v2f av; av.x = p[0]; av.y = p[1];
        a[mi] = av;
      }
      v2f b[2];
#pragma unroll
      for (int ni = 0; ni < 2; ++ni) {
        int col = wn * 32 + ni * 16 + ln;
        v2f bv;
        bv.x = lds_w[buf][kk + 2 * hi][col];
        bv.y = lds_w[buf][kk + 2 * hi + 1][col];
        b[ni] = bv;
      }
#pragma unroll
      for (int mi = 0; mi < 2; ++mi)
#pragma unroll
        for (int ni = 0; ni < 2; ++ni)
          acc[mi][ni] = __builtin_amdgcn_wmma_f32_16x16x4_f32(
              false, a[mi], false, b[ni], (short)0, acc[mi][ni], false, false);
    }
    __syncthreads();  // buffer reusable for chunk c+2 DMA
  }

  // ---- epilogue: out = x2 + w2 - 2*acc ----
#pragma unroll
  for (int mi = 0; mi < 2; ++mi) {
#pragma unroll
    for (int ni = 0; ni < 2; ++ni) {
      int colg = blockN + wn * 32 + ni * 16 + ln;
      float w2v = w2g[colg];
#pragma unroll
      for (int i = 0; i < 8; ++i) {
        int rowg = blockM + wm * 32 + mi * 16 + i + 8 * hi;
        out[(size_t)rowg * UNITS + colg] =
            x2g[rowg] + w2v - 2.0f * acc[mi][ni][i];
      }
    }
  }
}

// ---------------------------------------------------------------------------
extern "C" void kernel_launch(void* const* d_in, const int* in_sizes, int n_in,
                              void* d_out, int out_size, void* d_ws, size_t ws_size,
                              hipStream_t stream) {
  const float* x = (const float*)d_in[0];  // [32768, 128]
  const float* w = (const float*)d_in[1];  // [128, 1024]
  float* out = (float*)d_out;              // [32768, 1024]

  float* x2 = (float*)d_ws;                // 32768 fp32
  float* w2 = x2 + BN_ROWS;                // 1024 fp32

  row_norms_kernel<<<BN_ROWS / 256, 256, 0, stream>>>(x, x2);
  col_norms_kernel<<<UNITS / 256, 256, 0, stream>>>(w, w2);

  dim3 grid(UNITS / NB, BN_ROWS / MB);     // (16, 512)
  dist_gemm_kernel<<<grid, 128, 0, stream>>>(x, w, x2, w2, out);
}